// GraphUNet_4818953306737
// MI455X (gfx1250) — compile-verified
//
#include <hip/hip_runtime.h>
#include <cstdint>
#include <cstddef>

#define H      128
#define EDGES  800000
#define APAD   4
#define CDIV(a,b) (((a)+(b)-1)/(b))

typedef float v2f __attribute__((ext_vector_type(2)));
typedef float v8f __attribute__((ext_vector_type(8)));

// ---------------------------------------------------------------------------
// Elementwise utilities
// ---------------------------------------------------------------------------
__global__ void k_fill_i32(int* __restrict__ p, int v, int n) {
  int i = blockIdx.x * blockDim.x + threadIdx.x;
  if (i < n) p[i] = v;
}

__global__ void k_zero_f32(float* __restrict__ p, int n) {
  int i = blockIdx.x * blockDim.x + threadIdx.x;
  if (i < n) p[i] = 0.0f;
}

__global__ void k_relu(float* __restrict__ p, int n) {
  int i = blockIdx.x * blockDim.x + threadIdx.x;
  if (i < n) p[i] = fmaxf(p[i], 0.0f);
}

// deg currently holds sum of masked in-edge weights; add self-loop (+1) and rsqrt
__global__ void k_rsqrt1p(float* __restrict__ deg, int n) {
  int i = blockIdx.x * blockDim.x + threadIdx.x;
  if (i < n) deg[i] = rsqrtf(deg[i] + 1.0f);
}

// ---------------------------------------------------------------------------
// GCN: degree accumulation over masked edges
// ---------------------------------------------------------------------------
__global__ void k_edge_degree(const int* __restrict__ dst, const int* __restrict__ mask,
                              float* __restrict__ deg, int ne) {
  int e = blockIdx.x * blockDim.x + threadIdx.x;
  if (e < ne && mask[e]) atomicAdd(&deg[dst[e]], 1.0f);
}

// ---------------------------------------------------------------------------
// Dense matmul on the matrix pipe: Y[n x 128] = X[n x K] @ W[K x 128]
// - 64-row X strip staged to LDS via async global->LDS DMA (ASYNCcnt),
//   zero-filled tail rows -> branch-free compute loop.
// - Workgroup = 8 waves; wave w owns N-tile w; 4 M-subtiles per wave so each
//   B fragment is reused by 4 WMMAs (B traffic /4, 4-deep WMMA bursts).
// - V_WMMA_F32_16X16X4_F32, f32 fragment layouts per CDNA5 ISA 7.12.2.
// ---------------------------------------------------------------------------
template <int K>
__global__ void k_matmul_wmma(const float* __restrict__ X, const float* __restrict__ W,
                              float* __restrict__ Y, int n) {
  constexpr int KP = K + APAD;               // padded LDS row (breaks bank stride)
  constexpr int MT = 64;                     // rows per block
  __shared__ float As[MT * KP];

  const int m0  = blockIdx.x * MT;
  const int tid = threadIdx.x;

  // ---- stage X[m0 .. m0+63, 0..K) (one contiguous region) into LDS ----
  constexpr int CHUNKS = (MT * K * 4) / (256 * 16);   // 16B per thread per chunk
  const unsigned lbase = (unsigned)(uintptr_t)(&As[0]);   // low 32 bits == LDS offset
  const uint64_t gbase = (uint64_t)(uintptr_t)(X + (size_t)m0 * K);
  #pragma unroll
  for (int c = 0; c < CHUNKS; ++c) {
    const int byteoff = (tid + c * 256) * 16;
    const int row = byteoff / (K * 4);
    const int col = byteoff % (K * 4);
    if (m0 + row < n) {
      const unsigned laddr = lbase + (unsigned)(row * KP * 4 + col);
      const uint64_t gaddr = gbase + (uint64_t)byteoff;
      asm volatile("global_load_async_to_lds_b128 %0, %1, off"
                   :: "v"(laddr), "v"(gaddr) : "memory");
    } else {
      float* zp = &As[row * KP + (col >> 2)];
      zp[0] = 0.0f; zp[1] = 0.0f; zp[2] = 0.0f; zp[3] = 0.0f;
    }
  }
  asm volatile("s_wait_asynccnt 0x0" ::: "memory");
  __syncthreads();

  // ---- compute ----
  const int wave = tid >> 5;
  const int lane = tid & 31;
  const int n0   = wave * 16;                // 8 waves cover 128 output columns
  const int half = lane >> 4;                // 0: lanes 0-15, 1: lanes 16-31
  const int l15  = lane & 15;

  v8f acc[4] = {};
  #pragma unroll 4
  for (int k0 = 0; k0 < K; k0 += 4) {
    const int ks = k0 + half * 2;
    v2f b;                                   // B: K=ks/ks+1, N=l15 (per half)
    b.x = W[ks * H + n0 + l15];
    b.y = W[(ks + 1) * H + n0 + l15];
    #pragma unroll
    for (int mt = 0; mt < 4; ++mt) {         // A: M=l15 (+16*mt), K=ks/ks+1
      const v2f a = *(const v2f*)&As[(mt * 16 + l15) * KP + ks];
      acc[mt] = __builtin_amdgcn_wmma_f32_16x16x4_f32(false, a, false, b,
                                                      (short)0, acc[mt], false, false);
    }
  }

  // ---- store (uniform fast path for fully in-range blocks) ----
  const int colY = n0 + l15;
  if (m0 + MT <= n) {
    #pragma unroll
    for (int mt = 0; mt < 4; ++mt)
      #pragma unroll
      for (int r = 0; r < 8; ++r)            // D: VGPR r -> M = mt*16 + r + 8*half
        Y[(size_t)(m0 + mt * 16 + r + half * 8) * H + colY] = acc[mt][r];
  } else {
    #pragma unroll
    for (int mt = 0; mt < 4; ++mt)
      #pragma unroll
      for (int r = 0; r < 8; ++r) {
        const int row = m0 + mt * 16 + r + half * 8;
        if (row < n) Y[(size_t)row * H + colY] = acc[mt][r];
      }
  }
}

// ---------------------------------------------------------------------------
// GCN: self-loop term + bias:  y[v,f] = h[v,f]*dis[v]^2 + b[f]
// ---------------------------------------------------------------------------
__global__ void k_self_bias(const float* __restrict__ h, const float* __restrict__ dis,
                            const float* __restrict__ b, float* __restrict__ y, int n) {
  int i = blockIdx.x * blockDim.x + threadIdx.x;
  if (i < n * H) {
    int v = i >> 7, f = i & 127;
    float d = dis[v];
    y[i] = h[i] * d * d + b[f];
  }
}

// ---------------------------------------------------------------------------
// GCN: edge scatter, one wave32 per edge; 4 f32 atomics per lane (128 feats)
// ---------------------------------------------------------------------------
__global__ void k_edge_scatter(const float* __restrict__ h, const float* __restrict__ dis,
                               const int* __restrict__ src, const int* __restrict__ dst,
                               const int* __restrict__ mask, float* __restrict__ y, int ne) {
  int e    = blockIdx.x * (blockDim.x >> 5) + (threadIdx.x >> 5);
  int lane = threadIdx.x & 31;
  if (e >= ne || !mask[e]) return;
  int s = src[e], d = dst[e];
  float nrm = dis[s] * dis[d];
  const float* hs = h + (size_t)s * H;
  float*       yd = y + (size_t)d * H;
  #pragma unroll
  for (int u = 0; u < 4; ++u) {
    int f = lane + u * 32;
    atomicAdd(&yd[f], hs[f] * nrm);
  }
}

// ---------------------------------------------------------------------------
// GraphConv score: per-node dots with w_rel / w_root (wave32 per node)
// ---------------------------------------------------------------------------
__global__ void k_score_node(const float* __restrict__ x, const float* __restrict__ wrel,
                             const float* __restrict__ wroot, const float* __restrict__ b,
                             float* __restrict__ score, float* __restrict__ aval, int n) {
  int node = blockIdx.x * (blockDim.x >> 5) + (threadIdx.x >> 5);
  int lane = threadIdx.x & 31;
  if (node >= n) return;
  const float* xr = x + (size_t)node * H;
  float ar = 0.0f, rr = 0.0f;
  #pragma unroll
  for (int u = 0; u < 4; ++u) {
    int f = lane + u * 32;
    float v = xr[f];
    ar += v * wrel[f];
    rr += v * wroot[f];
  }
  #pragma unroll
  for (int off = 16; off > 0; off >>= 1) {
    ar += __shfl_xor(ar, off, 32);
    rr += __shfl_xor(rr, off, 32);
  }
  if (lane == 0) { aval[node] = ar; score[node] = rr + b[0]; }
}

__global__ void k_score_edge(const float* __restrict__ aval, const int* __restrict__ src,
                             const int* __restrict__ dst, const int* __restrict__ mask,
                             float* __restrict__ score, int ne) {
  int e = blockIdx.x * blockDim.x + threadIdx.x;
  if (e < ne && mask[e]) atomicAdd(&score[dst[e]], aval[src[e]]);
}

// ---------------------------------------------------------------------------
// Stable top-k via rank-by-count (matches lax.top_k: descending, ties->low idx)
// ---------------------------------------------------------------------------
#define RTILE 256
__global__ void k_rank(const float* __restrict__ s, int n, int* __restrict__ rank) {
  __shared__ float tile[RTILE];
  int i = blockIdx.x * blockDim.x + threadIdx.x;
  float si = (i < n) ? s[i] : 0.0f;
  int r = 0;
  for (int base = 0; base < n; base += RTILE) {
    int j = base + threadIdx.x;
    tile[threadIdx.x] = (j < n) ? s[j] : -__builtin_inff();
    __syncthreads();
    int lim = min(RTILE, n - base);
    for (int t = 0; t < lim; ++t) {
      float sj = tile[t];
      r += (sj > si) || (sj == si && (base + t) < i);
    }
    __syncthreads();
  }
  if (i < n) rank[i] = r;
}

__global__ void k_select(const int* __restrict__ rank, const float* __restrict__ score,
                         int n, int k, int* __restrict__ perm, float* __restrict__ sval) {
  int i = blockIdx.x * blockDim.x + threadIdx.x;
  if (i < n) {
    int r = rank[i];
    if (r < k) { perm[r] = i; sval[r] = score[i]; }
  }
}

// rev = argsort(perm): rev[rank_asc(perm[j])] = j   (perm values distinct)
__global__ void k_argsort_perm(const int* __restrict__ perm, int k, int* __restrict__ rev) {
  __shared__ int tile[RTILE];
  int j = blockIdx.x * blockDim.x + threadIdx.x;
  int pj = (j < k) ? perm[j] : 0;
  int r = 0;
  for (int base = 0; base < k; base += RTILE) {
    int t = base + threadIdx.x;
    tile[threadIdx.x] = (t < k) ? perm[t] : 0x7fffffff;
    __syncthreads();
    int lim = min(RTILE, k - base);
    for (int t2 = 0; t2 < lim; ++t2) r += (tile[t2] < pj);
    __syncthreads();
  }
  if (j < k) rev[r] = j;
}

// ---------------------------------------------------------------------------
// Pool / unpool helpers
// ---------------------------------------------------------------------------
__global__ void k_scatter_map(const int* __restrict__ perm, int* __restrict__ mapping, int k) {
  int j = blockIdx.x * blockDim.x + threadIdx.x;
  if (j < k) mapping[perm[j]] = j;
}

__global__ void k_filter_edges(int* __restrict__ src, int* __restrict__ dst,
                               int* __restrict__ mask, const int* __restrict__ mapping, int ne) {
  int e = blockIdx.x * blockDim.x + threadIdx.x;
  if (e >= ne) return;
  int ns = mapping[src[e]], nd = mapping[dst[e]];
  int nm = mask[e] && (ns >= 0) && (nd >= 0);
  src[e] = nm ? ns : 0;
  dst[e] = nm ? nd : 0;
  mask[e] = nm;
}

__global__ void k_filter_edges_rev(int* __restrict__ src, int* __restrict__ dst,
                                   int* __restrict__ mask, const int* __restrict__ mapping,
                                   const int* __restrict__ rev, int ne) {
  int e = blockIdx.x * blockDim.x + threadIdx.x;
  if (e >= ne) return;
  int ns = mapping[src[e]], nd = mapping[dst[e]];
  int nm = mask[e] && (ns >= 0) && (nd >= 0);
  src[e] = nm ? rev[ns] : 0;
  dst[e] = nm ? rev[nd] : 0;
  mask[e] = nm;
}

__global__ void k_pool_gather(const float* __restrict__ x, const int* __restrict__ perm,
                              const float* __restrict__ sval, float* __restrict__ xo, int k) {
  int i = blockIdx.x * blockDim.x + threadIdx.x;
  if (i < k * H) {
    int j = i >> 7, f = i & 127;
    xo[i] = x[(size_t)perm[j] * H + f] * tanhf(sval[j]);
  }
}

__global__ void k_unpool_concat(const float* __restrict__ x, const int* __restrict__ perm,
                                const float* __restrict__ skip, float* __restrict__ xc, int k) {
  int i = blockIdx.x * blockDim.x + threadIdx.x;
  if (i < k * 2 * H) {
    int j = i / (2 * H), c = i % (2 * H);
    xc[i] = (c < H) ? x[(size_t)perm[j] * H + c] : skip[(size_t)j * H + (c - H)];
  }
}

// ---------------------------------------------------------------------------
// Host-side stage helpers (all launches on `stream`)
// ---------------------------------------------------------------------------
static void run_gcn(const float* x, const float* W, const float* b, int n, int K,
                    const int* src, const int* dst, const int* mask,
                    float* hbuf, float* deg, float* y, hipStream_t stream) {
  k_zero_f32    <<<CDIV(n, 256), 256, 0, stream>>>(deg, n);
  k_edge_degree <<<CDIV(EDGES, 256), 256, 0, stream>>>(dst, mask, deg, EDGES);
  k_rsqrt1p     <<<CDIV(n, 256), 256, 0, stream>>>(deg, n);
  if (K == 128)
    k_matmul_wmma<128><<<CDIV(n, 64), 256, 0, stream>>>(x, W, hbuf, n);
  else
    k_matmul_wmma<256><<<CDIV(n, 64), 256, 0, stream>>>(x, W, hbuf, n);
  k_self_bias   <<<CDIV(n * H, 256), 256, 0, stream>>>(hbuf, deg, b, y, n);
  k_edge_scatter<<<CDIV(EDGES, 8), 256, 0, stream>>>(hbuf, deg, src, dst, mask, y, EDGES);
  k_relu        <<<CDIV(n * H, 256), 256, 0, stream>>>(y, n * H);
}

static void run_score(const float* x, const float* wrel, const float* wroot, const float* b,
                      const int* src, const int* dst, const int* mask, int n,
                      float* score, float* aval, hipStream_t stream) {
  k_score_node<<<CDIV(n, 8), 256, 0, stream>>>(x, wrel, wroot, b, score, aval, n);
  k_score_edge<<<CDIV(EDGES, 256), 256, 0, stream>>>(aval, src, dst, mask, score, EDGES);
}

static void run_topk(const float* score, int n, int k, int* rankA, int* perm, float* sval,
                     hipStream_t stream) {
  k_rank  <<<CDIV(n, 256), 256, 0, stream>>>(score, n, rankA);
  k_select<<<CDIV(n, 256), 256, 0, stream>>>(rankA, score, n, k, perm, sval);
}

// ---------------------------------------------------------------------------
// Orchestration
// ---------------------------------------------------------------------------
extern "C" void kernel_launch(void* const* d_in, const int* in_sizes, int n_in,
                              void* d_out, int out_size, void* d_ws, size_t ws_size,
                              hipStream_t stream) {
  (void)in_sizes; (void)n_in; (void)out_size; (void)ws_size;

  const float* x_in  = (const float*)d_in[0];
  const int*   e_src = (const int*)d_in[1];
  const int*   e_dst = (const int*)d_in[2];
  const float* dw[3] = {(const float*)d_in[3], (const float*)d_in[5], (const float*)d_in[7]};
  const float* db[3] = {(const float*)d_in[4], (const float*)d_in[6], (const float*)d_in[8]};
  const float* p_wrel[2]  = {(const float*)d_in[9],  (const float*)d_in[12]};
  const float* p_wroot[2] = {(const float*)d_in[10], (const float*)d_in[13]};
  const float* p_b[2]     = {(const float*)d_in[11], (const float*)d_in[14]};
  const float* u_wrel[2]  = {(const float*)d_in[15], (const float*)d_in[18]};
  const float* u_wroot[2] = {(const float*)d_in[16], (const float*)d_in[19]};
  const float* u_b[2]     = {(const float*)d_in[17], (const float*)d_in[20]};
  const float* uw[2] = {(const float*)d_in[21], (const float*)d_in[23]};
  const float* ub[2] = {(const float*)d_in[22], (const float*)d_in[24]};

  const int SIZES[3] = {50000, 25000, 12500};
  const int UP_KS[2] = {6250, 3125};

  // ---- workspace carve-out (256B aligned) ----
  size_t off = 0;
  auto carve = [&](size_t bytes) -> void* {
    void* p = (char*)d_ws + off;
    off += (bytes + 255) & ~(size_t)255;
    return p;
  };
  float* xbuf  = (float*)carve((size_t)50000 * H * 4);
  float* hbuf  = (float*)carve((size_t)50000 * H * 4);
  float* ybuf  = (float*)carve((size_t)50000 * H * 4);
  float* skip1 = (float*)carve((size_t)25000 * H * 4);
  float* skip2 = (float*)carve((size_t)12500 * H * 4);
  float* xcat  = (float*)carve((size_t)6250 * 2 * H * 4);
  int*   src   = (int*)carve((size_t)EDGES * 4);
  int*   dst   = (int*)carve((size_t)EDGES * 4);
  int*   mask  = (int*)carve((size_t)EDGES * 4);
  float* deg   = (float*)carve((size_t)50000 * 4);
  float* score = (float*)carve((size_t)50000 * 4);
  float* aval  = (float*)carve((size_t)50000 * 4);
  int*   rankA = (int*)carve((size_t)50000 * 4);
  int*   perm  = (int*)carve((size_t)25000 * 4);
  float* sval  = (float*)carve((size_t)25000 * 4);
  int*   mapn  = (int*)carve((size_t)50000 * 4);
  int*   rev   = (int*)carve((size_t)6250 * 4);

  // ---- init: copy inputs into mutable workspace (never touch d_in) ----
  hipMemcpyAsync(xbuf, x_in, (size_t)50000 * H * 4, hipMemcpyDeviceToDevice, stream);
  hipMemcpyAsync(src, e_src, (size_t)EDGES * 4, hipMemcpyDeviceToDevice, stream);
  hipMemcpyAsync(dst, e_dst, (size_t)EDGES * 4, hipMemcpyDeviceToDevice, stream);
  k_fill_i32<<<CDIV(EDGES, 256), 256, 0, stream>>>(mask, 1, EDGES);

  // ---- down path ----
  for (int i = 0; i < 3; ++i) {
    int n = SIZES[i];
    run_gcn(xbuf, dw[i], db[i], n, H, src, dst, mask, hbuf, deg, ybuf, stream);
    if (i == 1) hipMemcpyAsync(skip1, ybuf, (size_t)25000 * H * 4, hipMemcpyDeviceToDevice, stream);
    if (i == 2) hipMemcpyAsync(skip2, ybuf, (size_t)12500 * H * 4, hipMemcpyDeviceToDevice, stream);
    if (i != 2) {
      int k = SIZES[i + 1];
      run_score(ybuf, p_wrel[i], p_wroot[i], p_b[i], src, dst, mask, n, score, aval, stream);
      run_topk(score, n, k, rankA, perm, sval, stream);
      k_pool_gather<<<CDIV(k * H, 256), 256, 0, stream>>>(ybuf, perm, sval, xbuf, k);
      k_fill_i32   <<<CDIV(n, 256), 256, 0, stream>>>(mapn, -1, n);
      k_scatter_map<<<CDIV(k, 256), 256, 0, stream>>>(perm, mapn, k);
      k_filter_edges<<<CDIV(EDGES, 256), 256, 0, stream>>>(src, dst, mask, mapn, EDGES);
    }
  }
  // current x lives in ybuf (12500 x 128)

  // ---- up path ----
  float* xcur = ybuf;
  float* xnext = xbuf;
  int n = SIZES[2];
  const float* skips_up[2] = {skip2, skip1};
  for (int i = 0; i < 2; ++i) {
    int k = UP_KS[i];
    run_score(xcur, u_wrel[i], u_wroot[i], u_b[i], src, dst, mask, n, score, aval, stream);
    run_topk(score, n, k, rankA, perm, sval, stream);
    k_fill_i32    <<<CDIV(n, 256), 256, 0, stream>>>(mapn, -1, n);
    k_scatter_map <<<CDIV(k, 256), 256, 0, stream>>>(perm, mapn, k);
    k_argsort_perm<<<CDIV(k, 256), 256, 0, stream>>>(perm, k, rev);
    k_filter_edges_rev<<<CDIV(EDGES, 256), 256, 0, stream>>>(src, dst, mask, mapn, rev, EDGES);
    k_unpool_concat<<<CDIV(k * 2 * H, 256), 256, 0, stream>>>(xcur, perm, skips_up[i], xcat, k);
    run_gcn(xcat, uw[i], ub[i], k, 2 * H, src, dst, mask, hbuf, deg, xnext, stream);
    float* t = xcur; xcur = xnext; xnext = t;
    n = k;
  }

  // ---- output: final x is 3125 x 128 ----
  hipMemcpyAsync(d_out, xcur, (size_t)3125 * H * 4, hipMemcpyDeviceToDevice, stream);
}